// DSSnetwork_12352325943916
// MI455X (gfx1250) — compile-verified
//
#include <hip/hip_runtime.h>

typedef __attribute__((ext_vector_type(2))) float v2f;
typedef __attribute__((ext_vector_type(8))) float v8f;

#define EMBD 128

// ---------------------------------------------------------------- utilities
__global__ void dss_zero_f32(float* __restrict__ p, long n) {
  long i = (long)blockIdx.x * blockDim.x + threadIdx.x;
  long stride = (long)gridDim.x * blockDim.x;
  for (; i < n; i += stride) p[i] = 0.0f;
}

__global__ void dss_cumsum(const int* __restrict__ nnps, int* __restrict__ csum, int Gn) {
  if (blockIdx.x == 0 && threadIdx.x == 0) {
    int s = 0;
    for (int i = 0; i < Gn; ++i) { csum[i] = s; s += nnps[i]; }
    csum[Gn] = s;
  }
}

__global__ void dss_node_idx(const int* __restrict__ batch, const int* __restrict__ subidx,
                             const int* __restrict__ csum, int* __restrict__ nidx, int T) {
  int t = blockIdx.x * blockDim.x + threadIdx.x;
  if (t < T) nidx[t] = csum[batch[t]] + subidx[t];
}

// agg[dst] += x[src] over edges; one wave per edge, float4 per lane (128 cols)
__global__ void dss_scatter_edges(const float* __restrict__ x, const int* __restrict__ ei,
                                  float* __restrict__ agg, int E) {
  int gid = blockIdx.x * blockDim.x + threadIdx.x;
  int wid = gid >> 5, lane = gid & 31;
  int nw = (gridDim.x * blockDim.x) >> 5;
  for (int e = wid; e < E; e += nw) {
    int src = ei[e], dst = ei[E + e];
    float4 v = ((const float4*)(x + (long)src * EMBD))[lane];
    float* d = agg + (long)dst * EMBD + lane * 4;
    atomicAdd(d + 0, v.x); atomicAdd(d + 1, v.y);
    atomicAdd(d + 2, v.z); atomicAdd(d + 3, v.w);
  }
}

// out[seg[r]] += X[r]; cnt[seg[r]] += 1 (one wave per row)
__global__ void dss_seg_accum(const float* __restrict__ X, const int* __restrict__ seg,
                              float* __restrict__ out, float* __restrict__ cnt, int R) {
  int gid = blockIdx.x * blockDim.x + threadIdx.x;
  int wid = gid >> 5, lane = gid & 31;
  int nw = (gridDim.x * blockDim.x) >> 5;
  for (int r = wid; r < R; r += nw) {
    int s = seg[r];
    float4 v = ((const float4*)(X + (long)r * EMBD))[lane];
    float* d = out + (long)s * EMBD + lane * 4;
    atomicAdd(d + 0, v.x); atomicAdd(d + 1, v.y);
    atomicAdd(d + 2, v.z); atomicAdd(d + 3, v.w);
    if (lane == 0) atomicAdd(cnt + s, 1.0f);
  }
}

__global__ void dss_seg_div(float* __restrict__ M, const float* __restrict__ cnt, long total) {
  long i = (long)blockIdx.x * blockDim.x + threadIdx.x;
  if (i < total) M[i] = M[i] / fmaxf(cnt[i >> 7], 1.0f);
}

// ------------------------------------------------- fused dual GEMM via WMMA
// C[M x 128] = A1 @ B1 + A2 @ B2 + bias   (K = 128 each, M % 128 == 0)
// block = 256 threads (8 wave32); block computes a 128x128 tile; wave w owns
// rows w*16..w*16+15 across all 128 columns (8 accumulators of v8f).
// B is staged K-pair-interleaved in LDS so both fragment elements are one
// aligned 8-byte LDS load (ds_load_b64) straight into an even VGPR pair.
__global__ __launch_bounds__(256) void dss_gemm_dual_wmma(
    const float* __restrict__ A1, const float* __restrict__ B1,
    const float* __restrict__ A2, const float* __restrict__ B2,
    const float* __restrict__ bias, float* __restrict__ C) {
  __shared__ float lA[128 * 32];     // [row][k]            (row-major, 32 K)
  __shared__ float lB[16 * 256];     // [k/2][n*2 + (k&1)]  (K-pair interleaved)
  const int tid   = threadIdx.x;
  const int wave  = tid >> 5;
  const int lane  = tid & 31;
  const int lhalf = lane >> 4;   // 0: K pair {k,k+1}; 1: {k+2,k+3}
  const int llo   = lane & 15;
  const long rowBase = (long)blockIdx.x * 128;

  v8f acc[8] = {};

  for (int half = 0; half < 2; ++half) {
    const float* A = half ? A2 : A1;
    const float* B = half ? B2 : B1;
    for (int kc = 0; kc < 128; kc += 32) {
      __syncthreads();
#pragma unroll
      for (int i = 0; i < 4; ++i) {           // A chunk: 128 rows x 32 K
        int idx = tid + i * 256;              // float4 units, 0..1023
        int r = idx >> 3, c4 = idx & 7;
        *(float4*)(lA + r * 32 + c4 * 4) =
            *(const float4*)(A + (rowBase + r) * EMBD + kc + c4 * 4);
      }
#pragma unroll
      for (int i = 0; i < 4; ++i) {           // B chunk: 32 K x 128 N
        int idx = tid + i * 256;
        int k = idx >> 5, c4 = idx & 31;      // k row 0..31, n group 0..31
        float4 v = *(const float4*)(B + (long)(kc + k) * EMBD + c4 * 4);
        float* dst = lB + (k >> 1) * 256 + (k & 1);
        int n = c4 * 4;
        dst[(n + 0) * 2] = v.x;
        dst[(n + 1) * 2] = v.y;
        dst[(n + 2) * 2] = v.z;
        dst[(n + 3) * 2] = v.w;
      }
      __syncthreads();
#pragma unroll
      for (int k = 0; k < 32; k += 4) {
        // A fragment: lane row = llo, K = k + lhalf*2 (+0,+1), 8B aligned
        v2f afrag = *(const v2f*)(lA + (wave * 16 + llo) * 32 + k + lhalf * 2);
#pragma unroll
        for (int nt = 0; nt < 8; ++nt) {
          // B fragment: col = nt*16+llo, K pair (k/2 + lhalf), contiguous 8B
          v2f bfrag = *(const v2f*)(lB + (k / 2 + lhalf) * 256 +
                                    (nt * 16 + llo) * 2);
          acc[nt] = __builtin_amdgcn_wmma_f32_16x16x4_f32(
              false, afrag, false, bfrag, (short)0, acc[nt], false, false);
        }
      }
    }
  }
  __syncthreads();
  // C/D layout: lanes 0-15 -> N=lane, VGPR r -> M=r; lanes 16-31 -> M=r+8
  const long row0 = rowBase + wave * 16 + lhalf * 8;
#pragma unroll
  for (int nt = 0; nt < 8; ++nt) {
    int col = nt * 16 + llo;
    float bv = bias[col];
#pragma unroll
    for (int r = 0; r < 8; ++r)
      C[(row0 + r) * EMBD + col] = acc[nt][r] + bv;
  }
}

// ----------------------------------------------------------- BN statistics
__global__ void dss_col_stats(const float* __restrict__ M, long R,
                              float* __restrict__ sum, float* __restrict__ sq) {
  __shared__ float ls[256], lq[256];
  int col = threadIdx.x & 127;
  int sub = threadIdx.x >> 7;  // 0..1
  float s = 0.0f, q = 0.0f;
  for (long r = (long)blockIdx.x * 2 + sub; r < R; r += (long)gridDim.x * 2) {
    float v = M[r * EMBD + col];
    s += v; q += v * v;
  }
  ls[threadIdx.x] = s; lq[threadIdx.x] = q;
  __syncthreads();
  if (sub == 0) {
    atomicAdd(sum + col, ls[col] + ls[col + 128]);
    atomicAdd(sq  + col, lq[col] + lq[col + 128]);
  }
}

__global__ void dss_bn_finalize(const float* __restrict__ sum1, const float* __restrict__ sq1, float R1,
                                const float* __restrict__ sum2, const float* __restrict__ sq2, float R2,
                                float* __restrict__ mu1, float* __restrict__ rstd1,
                                float* __restrict__ mu2, float* __restrict__ rstd2) {
  int c = threadIdx.x;
  if (c < 128) {
    float m = sum1[c] / R1;
    mu1[c] = m;
    rstd1[c] = rsqrtf(sq1[c] / R1 - m * m + 1e-5f);
    m = sum2[c] / R2;
    mu2[c] = m;
    rstd2[c] = rsqrtf(sq2[c] / R2 - m * m + 1e-5f);
  }
}

// x_out = relu( BN1(h1) + BN2(h2)[node_idx] )
__global__ void dss_combine(const float* __restrict__ h1, const float* __restrict__ h2,
                            const int* __restrict__ nidx,
                            const float* __restrict__ mu1, const float* __restrict__ rstd1,
                            const float* __restrict__ g1, const float* __restrict__ be1,
                            const float* __restrict__ mu2, const float* __restrict__ rstd2,
                            const float* __restrict__ g2, const float* __restrict__ be2,
                            float* __restrict__ xout, long total) {
  long i = (long)blockIdx.x * blockDim.x + threadIdx.x;
  long stride = (long)gridDim.x * blockDim.x;
  for (; i < total; i += stride) {
    int c = (int)(i & 127);
    long t = i >> 7;
    int s = nidx[t];
    float a = (h1[i] - mu1[c]) * rstd1[c] * g1[c] + be1[c];
    float b = (h2[(long)s * EMBD + c] - mu2[c]) * rstd2[c] * g2[c] + be2[c];
    xout[i] = fmaxf(a + b, 0.0f);
  }
}

// --------------------------------------------------------------- MLP head
// out[16 x 10] = relu(hg[16 x 128] @ Wf1[128 x 256] + bf1) @ Wf2[256 x 10] + bf2
__global__ __launch_bounds__(256) void dss_head(const float* __restrict__ hg,
                                                const float* __restrict__ Wf1,
                                                const float* __restrict__ bf1,
                                                const float* __restrict__ Wf2,
                                                const float* __restrict__ bf2,
                                                float* __restrict__ out, int Gn) {
  __shared__ float sg[16 * 128];
  __shared__ float sh[16 * 256];
  int tid = threadIdx.x;
  for (int i = tid; i < Gn * 128; i += 256) sg[i] = hg[i];
  __syncthreads();
  for (int i = tid; i < Gn * 256; i += 256) {
    int g = i >> 8, j = i & 255;
    float s = bf1[j];
    for (int k = 0; k < 128; ++k) s += sg[g * 128 + k] * Wf1[k * 256 + j];
    sh[i] = fmaxf(s, 0.0f);
  }
  __syncthreads();
  for (int i = tid; i < Gn * 10; i += 256) {
    int g = i / 10, t = i % 10;
    float s = bf2[t];
    for (int k = 0; k < 256; ++k) s += sh[g * 256 + k] * Wf2[k * 10 + t];
    out[g * 10 + t] = s;
  }
}

// ----------------------------------------------------------------- driver
extern "C" void kernel_launch(void* const* d_in, const int* in_sizes, int n_in,
                              void* d_out, int out_size, void* d_ws, size_t ws_size,
                              hipStream_t stream) {
  const float* x0    = (const float*)d_in[0];
  const float* Wr    = (const float*)d_in[1];
  const float* Wn    = (const float*)d_in[2];
  const float* bL    = (const float*)d_in[3];
  const float* gam   = (const float*)d_in[4];
  const float* bet   = (const float*)d_in[5];
  const float* Wr_s  = (const float*)d_in[6];
  const float* Wn_s  = (const float*)d_in[7];
  const float* b_sL  = (const float*)d_in[8];
  const float* gam_s = (const float*)d_in[9];
  const float* bet_s = (const float*)d_in[10];
  const float* Wf1   = (const float*)d_in[11];
  const float* bf1   = (const float*)d_in[12];
  const float* Wf2   = (const float*)d_in[13];
  const float* bf2   = (const float*)d_in[14];
  const int* ei      = (const int*)d_in[15];
  const int* oei     = (const int*)d_in[16];
  const int* batch   = (const int*)d_in[17];
  const int* subidx  = (const int*)d_in[18];
  const int* subbat  = (const int*)d_in[19];
  const int* nnps    = (const int*)d_in[20];
  const int* sidxb   = (const int*)d_in[21];

  const int T  = in_sizes[0] / EMBD;      // 262144
  const int E  = in_sizes[15] / 2;        // 2097152
  const int EO = in_sizes[16] / 2;        // 32768
  const int Gn = in_sizes[20];            // 16
  const int S  = in_sizes[21];            // 2048
  const int L  = in_sizes[3] / EMBD;      // 4

  // ---- workspace carve-out (256B aligned)
  char* w = (char*)d_ws;
  size_t off = 0;
  auto alloc = [&](size_t bytes) -> char* {
    char* p = w + off;
    off += (bytes + 255) & ~(size_t)255;
    return p;
  };
  float* x_cur  = (float*)alloc((size_t)T * EMBD * 4);
  float* agg    = (float*)alloc((size_t)T * EMBD * 4);
  float* h1     = (float*)alloc((size_t)T * EMBD * 4);
  float* xs     = (float*)alloc((size_t)S * EMBD * 4 + (size_t)S * 4);
  float* xcnt   = xs + (size_t)S * EMBD;
  float* aggs   = (float*)alloc((size_t)S * EMBD * 4);
  float* h2     = (float*)alloc((size_t)S * EMBD * 4);
  float* stats  = (float*)alloc(8 * 128 * 4);
  float* sum1 = stats,        *sq1  = stats + 128;
  float* sum2 = stats + 256,  *sq2  = stats + 384;
  float* mu1  = stats + 512,  *rst1 = stats + 640;
  float* mu2  = stats + 768,  *rst2 = stats + 896;
  float* pool_s = (float*)alloc((size_t)S * EMBD * 4 + (size_t)S * 4);
  float* cnt_s  = pool_s + (size_t)S * EMBD;
  float* pool_g = (float*)alloc((size_t)Gn * EMBD * 4 + (size_t)Gn * 4);
  float* cnt_g  = pool_g + (size_t)Gn * EMBD;
  int* csum = (int*)alloc((Gn + 1) * 4);
  int* nidx = (int*)alloc((size_t)T * 4);

  // ---- node_idx = cumsum(num_nodes_per_subgraph)[batch] + subgraph_node_idx
  dss_cumsum<<<1, 32, 0, stream>>>(nnps, csum, Gn);
  dss_node_idx<<<(T + 255) / 256, 256, 0, stream>>>(batch, subidx, csum, nidx, T);

  const long TN = (long)T * EMBD;
  const long SN = (long)S * EMBD;

  for (int l = 0; l < L; ++l) {
    const float* xin = (l == 0) ? x0 : x_cur;
    // --- node-level neighbor aggregation
    dss_zero_f32<<<4096, 256, 0, stream>>>(agg, TN);
    dss_scatter_edges<<<4096, 256, 0, stream>>>(xin, ei, agg, E);
    // --- subgraph scatter-mean x_sum
    dss_zero_f32<<<1024, 256, 0, stream>>>(xs, SN + S);
    dss_seg_accum<<<4096, 256, 0, stream>>>(xin, nidx, xs, xcnt, T);
    dss_seg_div<<<(int)((SN + 255) / 256), 256, 0, stream>>>(xs, xcnt, SN);
    // --- big fused GEMM: h1 = x@Wr + agg@Wn + b   (WMMA f32 16x16x4)
    dss_gemm_dual_wmma<<<T / 128, 256, 0, stream>>>(
        xin, Wr + (size_t)l * EMBD * EMBD, agg, Wn + (size_t)l * EMBD * EMBD,
        bL + (size_t)l * EMBD, h1);
    // --- original-graph aggregation + small fused GEMM
    dss_zero_f32<<<1024, 256, 0, stream>>>(aggs, SN);
    dss_scatter_edges<<<1024, 256, 0, stream>>>(xs, oei, aggs, EO);
    dss_gemm_dual_wmma<<<S / 128, 256, 0, stream>>>(
        xs, Wr_s + (size_t)l * EMBD * EMBD, aggs, Wn_s + (size_t)l * EMBD * EMBD,
        b_sL + (size_t)l * EMBD, h2);
    // --- batch-norm statistics (biased, training mode)
    dss_zero_f32<<<2, 256, 0, stream>>>(stats, 512);
    dss_col_stats<<<2048, 256, 0, stream>>>(h1, T, sum1, sq1);
    dss_col_stats<<<64, 256, 0, stream>>>(h2, S, sum2, sq2);
    dss_bn_finalize<<<1, 128, 0, stream>>>(sum1, sq1, (float)T, sum2, sq2, (float)S,
                                           mu1, rst1, mu2, rst2);
    // --- x = relu(BN1(h1) + BN2(h2)[node_idx])
    dss_combine<<<8192, 256, 0, stream>>>(
        h1, h2, nidx, mu1, rst1, gam + (size_t)l * EMBD, bet + (size_t)l * EMBD,
        mu2, rst2, gam_s + (size_t)l * EMBD, bet_s + (size_t)l * EMBD, x_cur, TN);
  }

  // ---- pooling: mean per subgraph, then mean per graph
  dss_zero_f32<<<1024, 256, 0, stream>>>(pool_s, SN + S);
  dss_seg_accum<<<4096, 256, 0, stream>>>(x_cur, subbat, pool_s, cnt_s, T);
  dss_seg_div<<<(int)((SN + 255) / 256), 256, 0, stream>>>(pool_s, cnt_s, SN);
  dss_zero_f32<<<16, 256, 0, stream>>>(pool_g, (long)Gn * EMBD + Gn);
  dss_seg_accum<<<64, 256, 0, stream>>>(pool_s, sidxb, pool_g, cnt_g, S);
  dss_seg_div<<<(int)(((long)Gn * EMBD + 255) / 256), 256, 0, stream>>>(
      pool_g, cnt_g, (long)Gn * EMBD);

  // ---- MLP head -> d_out (16 x 10 fp32)
  dss_head<<<1, 256, 0, stream>>>(pool_g, Wf1, bf1, Wf2, bf2, (float*)d_out, Gn);
}